// PointNetfeat_56908316672769
// MI455X (gfx1250) — compile-verified
//
#include <hip/hip_runtime.h>
#include <hip/hip_bf16.h>
#include <math.h>

// ---------------------------------------------------------------------------
// PointNet feature head for MI455X (gfx1250), fp32 WMMA path.
//   B=4, N=1024, K=520, channels 6 -> 64 -> 128 -> 1024, out (4,1024)
// All GEMMs use V_WMMA_F32_16X16X4_F32. BN stats are computed analytically
// (6x6 feat covariance for layer1; 128-vector mean + 128x128 Gram for layer3)
// so no huge intermediates are materialized. Workspace use ~5.3 MB.
// ---------------------------------------------------------------------------

typedef float v2f __attribute__((ext_vector_type(2)));
typedef float v8f __attribute__((ext_vector_type(8)));

#define BN_B     4
#define BN_N     1024
#define BN_K     520
#define CHST     ((size_t)BN_N * BN_K)      // per-channel stride in info
#define S_FEAT   (4.0f * 1024.0f * 520.0f)  // 2,129,920 samples for BN1
#define S_BN     4096.0f                    // B*N samples for BN2/BN3
#define EPS      1e-5f

// workspace layout (float offsets)
#define OFF_S1   0        // 6   feat channel sums
#define OFF_S2   8        // 21  feat second-moment triangle
#define OFF_WA   32       // 64x4 effective layer1 A (K padded to 4)
#define OFF_WD   288      // 64x4 (uses 3) delta weights for x-fold
#define OFF_B1   544      // 64  effective layer1 bias
#define OFF_SC2  608      // 128 BN2 scale
#define OFF_SH2  736      // 128 BN2 shift
#define OFF_MU2  864      // 128 mean of h2
#define OFF_SC3  1024     // 1024 BN3 scale
#define OFF_SH3  2048     // 1024 BN3 shift
#define OFF_G    3072     // 128x128 Gram of h2
#define OFF_H1   20480    // 64 x 4096   (channel-major)
#define OFF_H2R  282624   // 128 x 4096  raw layer2 output
#define OFF_H2   806912   // 128 x 4096  normalized+relu layer2 output
// total = 1,331,200 floats = 5.3 MB

__device__ __forceinline__ v8f wmma4(v2f a, v2f b, v8f c) {
  // D(16x16,f32) = A(16x4,f32) x B(4x16,f32) + C
  return __builtin_amdgcn_wmma_f32_16x16x4_f32(
      false, a, false, b, (short)0, c, false, false);
}

// ---------------------------------------------------------------------------
__global__ __launch_bounds__(32) void k_zero(float* wsf) {
  wsf[threadIdx.x] = 0.0f; // zero S1 + S2 region (32 floats)
}

// feat = [info - x (3ch), x (3ch)]; accumulate 6 sums + 21 second moments
__global__ __launch_bounds__(128) void k_feat_stats(const float* __restrict__ x,
                                                    const float* __restrict__ info,
                                                    float* __restrict__ wsf) {
  int bn = blockIdx.x, b = bn >> 10, n = bn & 1023;
  const float* ib = info + (size_t)b * 3 * CHST + (size_t)n * BN_K;
  float x0 = x[(b * 3 + 0) * 1024 + n];
  float x1 = x[(b * 3 + 1) * 1024 + n];
  float x2 = x[(b * 3 + 2) * 1024 + n];
  float s1[6] = {0, 0, 0, 0, 0, 0};
  float s2[21];
#pragma unroll
  for (int i = 0; i < 21; ++i) s2[i] = 0.0f;
  for (int k = threadIdx.x; k < BN_K; k += 128) {
    float f[6];
    f[0] = ib[k] - x0;
    f[1] = ib[CHST + k] - x1;
    f[2] = ib[2 * CHST + k] - x2;
    f[3] = x0; f[4] = x1; f[5] = x2;
    int idx = 0;
#pragma unroll
    for (int c = 0; c < 6; ++c) {
      s1[c] += f[c];
#pragma unroll
      for (int cp = c; cp < 6; ++cp) s2[idx++] += f[c] * f[cp];
    }
  }
  __shared__ float red[27][128];
  int t = threadIdx.x;
#pragma unroll
  for (int i = 0; i < 6; ++i) red[i][t] = s1[i];
#pragma unroll
  for (int i = 0; i < 21; ++i) red[6 + i][t] = s2[i];
  __syncthreads();
  for (int st = 64; st > 0; st >>= 1) {
    if (t < st)
      for (int i = 0; i < 27; ++i) red[i][t] += red[i][t + st];
    __syncthreads();
  }
  if (t == 0)
    for (int i = 0; i < 27; ++i)
      atomicAdd(&wsf[i < 6 ? OFF_S1 + i : OFF_S2 + (i - 6)], red[i][0]);
}

// fold BN1 + x-constant channels into effective 64x3 weights + biases
__global__ __launch_bounds__(64) void k_prep1(const float* __restrict__ w1,
                                              const float* __restrict__ b1,
                                              const float* __restrict__ g1,
                                              const float* __restrict__ be1,
                                              float* __restrict__ wsf) {
  int o = threadIdx.x;
  float mu[6];
#pragma unroll
  for (int c = 0; c < 6; ++c) mu[c] = wsf[OFF_S1 + c] / S_FEAT;
  float mean = b1[o];
#pragma unroll
  for (int c = 0; c < 6; ++c) mean += w1[o * 6 + c] * mu[c];
  float var = 0.0f;
  for (int c = 0; c < 6; ++c)
    for (int cp = 0; cp < 6; ++cp) {
      int lo = c < cp ? c : cp, hi = c < cp ? cp : c;
      int idx = 6 * lo - lo * (lo - 1) / 2 + (hi - lo);
      float cov = wsf[OFF_S2 + idx] / S_FEAT - mu[c] * mu[cp];
      var += w1[o * 6 + c] * w1[o * 6 + cp] * cov;
    }
  float s = g1[o] * rsqrtf(var + EPS);
#pragma unroll
  for (int c = 0; c < 3; ++c) {
    wsf[OFF_WA + o * 4 + c] = w1[o * 6 + c] * s;
    wsf[OFF_WD + o * 4 + c] = (w1[o * 6 + c + 3] - w1[o * 6 + c]) * s;
  }
  wsf[OFF_WA + o * 4 + 3] = 0.0f;
  wsf[OFF_B1 + o] = (b1[o] - mean) * s + be1[o];
}

// layer1: per (b,n): relu(W'(64x3) @ info(3x520) + bias(b,n,o)), max over K
// one block per (b,n), 4 waves, each wave owns one 16-row tile of output.
// Hot loop is branch-free: 32 full 16-col tiles, then one masked tail tile.
__global__ __launch_bounds__(128) void k_layer1(const float* __restrict__ x,
                                                const float* __restrict__ info,
                                                float* __restrict__ wsf) {
  int bn = blockIdx.x, b = bn >> 10, n = bn & 1023;
  int tid = threadIdx.x, wave = tid >> 5, lane = tid & 31;
  int half = lane >> 4, l = lane & 15;
  __shared__ float biasRow[64];
  if (tid < 64) {
    float acc = wsf[OFF_B1 + tid];
#pragma unroll
    for (int c = 0; c < 3; ++c)
      acc += wsf[OFF_WD + tid * 4 + c] * x[(b * 3 + c) * 1024 + n];
    biasRow[tid] = acc;
  }
  __syncthreads();
  int mbase = wave * 16;
  int arow = mbase + l;
  v2f a;
  a.x = wsf[OFF_WA + arow * 4 + (half ? 2 : 0)];
  a.y = wsf[OFF_WA + arow * 4 + (half ? 3 : 1)];
  float bsv[8], mx[8];
#pragma unroll
  for (int r = 0; r < 8; ++r) {
    bsv[r] = biasRow[mbase + r + half * 8];
    mx[r] = 0.0f; // relu output >= 0
  }
  const float* ib = info + (size_t)b * 3 * CHST + (size_t)n * BN_K;
  const float* ib0 = ib + (half ? 2 * CHST : 0); // K-row 0: ch0 (lo) / ch2 (hi)
  const float* ib1 = ib + CHST;                  // K-row 1: ch1 (lo) / 0 (hi)
  const float c1mask = half ? 0.0f : 1.0f;
  // 32 full tiles, branch-free
#pragma unroll 4
  for (int kt = 0; kt < 32; ++kt) {
    int col = kt * 16 + l;
    v2f bv;
    bv.x = ib0[col];
    bv.y = ib1[col] * c1mask;
    v8f d = {};
    d = wmma4(a, bv, d);
#pragma unroll
    for (int r = 0; r < 8; ++r)
      mx[r] = fmaxf(mx[r], fmaxf(d[r] + bsv[r], 0.0f));
  }
  // tail tile: cols 512..519 valid (l < 8)
  {
    int valid = l < 8;
    int cc = valid ? 512 + l : 519;
    v2f bv;
    bv.x = valid ? ib0[cc] : 0.0f;
    bv.y = valid ? ib1[cc] * c1mask : 0.0f;
    v8f d = {};
    d = wmma4(a, bv, d);
#pragma unroll
    for (int r = 0; r < 8; ++r) {
      float hv = fmaxf(d[r] + bsv[r], 0.0f);
      hv = valid ? hv : 0.0f;           // masked to 0: no-op for max (mx>=0)
      mx[r] = fmaxf(mx[r], hv);
    }
  }
#pragma unroll
  for (int r = 0; r < 8; ++r)
    for (int off = 1; off < 16; off <<= 1)
      mx[r] = fmaxf(mx[r], __shfl_xor(mx[r], off, 16));
  if (l == 0) {
    float* h1 = wsf + OFF_H1;
#pragma unroll
    for (int r = 0; r < 8; ++r)
      h1[(size_t)(mbase + r + half * 8) * 4096 + bn] = mx[r];
  }
}

// layer2 raw: h2raw(128x4096) = w2(128x64) @ h1(64x4096) + b2
__global__ __launch_bounds__(128) void k_layer2(const float* __restrict__ w2,
                                                const float* __restrict__ b2,
                                                float* __restrict__ wsf) {
  int wg = blockIdx.x * 4 + (threadIdx.x >> 5); // 0..2047 tiles
  int mt = wg & 7, nt = wg >> 3;
  int lane = threadIdx.x & 31, half = lane >> 4, l = lane & 15;
  int col = nt * 16 + l;
  int arow = mt * 16 + l;
  // hoist A fragments for all 16 K-steps
  v2f areg[16];
#pragma unroll
  for (int ks = 0; ks < 16; ++ks) {
    int c0 = ks * 4 + (half ? 2 : 0);
    areg[ks].x = w2[arow * 64 + c0];
    areg[ks].y = w2[arow * 64 + c0 + 1];
  }
  const float* h1 = wsf + OFF_H1;
  v8f acc = {};
#pragma unroll
  for (int ks = 0; ks < 16; ++ks) {
    int kb = ks * 4;
    v2f bv;
    bv.x = h1[(size_t)(kb + (half ? 2 : 0)) * 4096 + col];
    bv.y = h1[(size_t)(kb + (half ? 3 : 1)) * 4096 + col];
    acc = wmma4(areg[ks], bv, acc);
  }
  float* h2r = wsf + OFF_H2R;
#pragma unroll
  for (int r = 0; r < 8; ++r) {
    int row = mt * 16 + r + half * 8;
    h2r[(size_t)row * 4096 + col] = acc[r] + b2[row];
  }
}

// BN2 stats per channel over 4096 samples -> scale/shift
__global__ __launch_bounds__(256) void k_bnstats2(const float* __restrict__ g2,
                                                  const float* __restrict__ be2,
                                                  float* __restrict__ wsf) {
  int c = blockIdx.x;
  const float* h2r = wsf + OFF_H2R + (size_t)c * 4096;
  float s = 0.0f, s2 = 0.0f;
  for (int i = threadIdx.x; i < 4096; i += 256) {
    float v = h2r[i];
    s += v; s2 += v * v;
  }
  __shared__ float ra[256], rb[256];
  ra[threadIdx.x] = s; rb[threadIdx.x] = s2;
  __syncthreads();
  for (int st = 128; st > 0; st >>= 1) {
    if (threadIdx.x < st) { ra[threadIdx.x] += ra[threadIdx.x + st]; rb[threadIdx.x] += rb[threadIdx.x + st]; }
    __syncthreads();
  }
  if (threadIdx.x == 0) {
    float mean = ra[0] / S_BN;
    float var = rb[0] / S_BN - mean * mean;
    float sc = g2[c] * rsqrtf(var + EPS);
    wsf[OFF_SC2 + c] = sc;
    wsf[OFF_SH2 + c] = be2[c] - mean * sc;
  }
}

__global__ __launch_bounds__(512) void k_relunorm(float* __restrict__ wsf) {
  int idx = blockIdx.x * 512 + threadIdx.x; // 128*4096 total
  int c = idx >> 12;
  wsf[OFF_H2 + idx] = fmaxf(wsf[OFF_H2R + idx] * wsf[OFF_SC2 + c] + wsf[OFF_SH2 + c], 0.0f);
}

__global__ __launch_bounds__(256) void k_mu2(float* __restrict__ wsf) {
  int c = blockIdx.x;
  const float* h2 = wsf + OFF_H2 + (size_t)c * 4096;
  float s = 0.0f;
  for (int i = threadIdx.x; i < 4096; i += 256) s += h2[i];
  __shared__ float ra[256];
  ra[threadIdx.x] = s;
  __syncthreads();
  for (int st = 128; st > 0; st >>= 1) {
    if (threadIdx.x < st) ra[threadIdx.x] += ra[threadIdx.x + st];
    __syncthreads();
  }
  if (threadIdx.x == 0) wsf[OFF_MU2 + c] = ra[0] / S_BN;
}

// Gram of h2: G = h2 @ h2^T (128x128), for analytic BN3 variance
__global__ __launch_bounds__(128) void k_gram(float* __restrict__ wsf) {
  int i = blockIdx.x, j = threadIdx.x;
  const float* hi = wsf + OFF_H2 + (size_t)i * 4096;
  const float* hj = wsf + OFF_H2 + (size_t)j * 4096;
  float s = 0.0f;
  for (int k = 0; k < 4096; ++k) s += hi[k] * hj[k];
  wsf[OFF_G + i * 128 + j] = s;
}

// BN3 scale/shift per output channel: var = w3^T G w3 / S - (w3.mu2)^2
__global__ __launch_bounds__(256) void k_prep3(const float* __restrict__ w3,
                                               const float* __restrict__ b3,
                                               const float* __restrict__ g3,
                                               const float* __restrict__ be3,
                                               float* __restrict__ wsf) {
  int o = blockIdx.x * 256 + threadIdx.x; // 0..1023
  const float* wo = w3 + (size_t)o * 128;
  float d = 0.0f;
  for (int c = 0; c < 128; ++c) d += wo[c] * wsf[OFF_MU2 + c];
  float q = 0.0f;
  for (int c = 0; c < 128; ++c) {
    float t = 0.0f;
    const float* Gc = wsf + OFF_G + c * 128;
    for (int cp = 0; cp < 128; ++cp) t += Gc[cp] * wo[cp];
    q += wo[c] * t;
  }
  float var = q / S_BN - d * d;
  float mean = d + b3[o];
  float sc = g3[o] * rsqrtf(var + EPS);
  wsf[OFF_SC3 + o] = sc;
  wsf[OFF_SH3 + o] = be3[o] - mean * sc;
}

// layer3 fused: out[b,o] = bnorm( extremum_n( w3 @ h2 + b3 ) )
// tracks both max and min so a negative BN scale still commutes correctly.
// A fragments (one w3 row slice per lane, 32 K-steps) hoisted to registers.
__global__ __launch_bounds__(128) void k_layer3max(const float* __restrict__ w3,
                                                   const float* __restrict__ b3,
                                                   float* __restrict__ wsf,
                                                   float* __restrict__ out) {
  int b = blockIdx.x >> 4;
  int mt = (blockIdx.x & 15) * 4 + (threadIdx.x >> 5); // 0..63, rows mt*16..+15
  int lane = threadIdx.x & 31, half = lane >> 4, l = lane & 15;
  int arow = mt * 16 + l;
  v2f areg[32];
#pragma unroll
  for (int ks = 0; ks < 32; ++ks) {
    int c0 = ks * 4 + (half ? 2 : 0);
    areg[ks].x = w3[(size_t)arow * 128 + c0];
    areg[ks].y = w3[(size_t)arow * 128 + c0 + 1];
  }
  const float* h2 = wsf + OFF_H2;
  float b3v[8], rmax[8], rmin[8];
#pragma unroll
  for (int r = 0; r < 8; ++r) {
    b3v[r] = b3[mt * 16 + r + half * 8];
    rmax[r] = -3.402823466e38f;
    rmin[r] = 3.402823466e38f;
  }
  for (int nt = 0; nt < 64; ++nt) {
    int col = b * 1024 + nt * 16 + l;
    v8f acc = {};
#pragma unroll
    for (int ks = 0; ks < 32; ++ks) {
      int kb = ks * 4;
      v2f bv;
      bv.x = h2[(size_t)(kb + (half ? 2 : 0)) * 4096 + col];
      bv.y = h2[(size_t)(kb + (half ? 3 : 1)) * 4096 + col];
      acc = wmma4(areg[ks], bv, acc);
    }
#pragma unroll
    for (int r = 0; r < 8; ++r) {
      float v = acc[r] + b3v[r];
      rmax[r] = fmaxf(rmax[r], v);
      rmin[r] = fminf(rmin[r], v);
    }
  }
#pragma unroll
  for (int r = 0; r < 8; ++r)
    for (int off = 1; off < 16; off <<= 1) {
      rmax[r] = fmaxf(rmax[r], __shfl_xor(rmax[r], off, 16));
      rmin[r] = fminf(rmin[r], __shfl_xor(rmin[r], off, 16));
    }
  if (l == 0) {
#pragma unroll
    for (int r = 0; r < 8; ++r) {
      int o = mt * 16 + r + half * 8;
      float sc = wsf[OFF_SC3 + o], sh = wsf[OFF_SH3 + o];
      float raw = sc >= 0.0f ? rmax[r] : rmin[r];
      out[b * 1024 + o] = raw * sc + sh;
    }
  }
}

// ---------------------------------------------------------------------------
extern "C" void kernel_launch(void* const* d_in, const int* in_sizes, int n_in,
                              void* d_out, int out_size, void* d_ws, size_t ws_size,
                              hipStream_t stream) {
  const float* x    = (const float*)d_in[0];
  const float* info = (const float*)d_in[1];
  const float* w1   = (const float*)d_in[2];
  const float* b1   = (const float*)d_in[3];
  const float* g1   = (const float*)d_in[4];
  const float* be1  = (const float*)d_in[5];
  const float* w2   = (const float*)d_in[6];
  const float* b2   = (const float*)d_in[7];
  const float* g2   = (const float*)d_in[8];
  const float* be2  = (const float*)d_in[9];
  const float* w3   = (const float*)d_in[10];
  const float* b3   = (const float*)d_in[11];
  const float* g3   = (const float*)d_in[12];
  const float* be3  = (const float*)d_in[13];
  float* wsf = (float*)d_ws;
  float* out = (float*)d_out;

  k_zero<<<1, 32, 0, stream>>>(wsf);
  k_feat_stats<<<4096, 128, 0, stream>>>(x, info, wsf);
  k_prep1<<<1, 64, 0, stream>>>(w1, b1, g1, be1, wsf);
  k_layer1<<<4096, 128, 0, stream>>>(x, info, wsf);
  k_layer2<<<512, 128, 0, stream>>>(w2, b2, wsf);
  k_bnstats2<<<128, 256, 0, stream>>>(g2, be2, wsf);
  k_relunorm<<<1024, 512, 0, stream>>>(wsf);
  k_mu2<<<128, 256, 0, stream>>>(wsf);
  k_gram<<<128, 128, 0, stream>>>(wsf);
  k_prep3<<<4, 256, 0, stream>>>(w3, b3, g3, be3, wsf);
  k_layer3max<<<64, 128, 0, stream>>>(w3, b3, wsf, out);
}